// GatedConvSeqMixer_20744692039804
// MI455X (gfx1250) — compile-verified
//
#include <hip/hip_runtime.h>
#include <hip/hip_bf16.h>
#include <math.h>

// Problem constants (fixed by the reference).
constexpr int Bc = 8, Lc = 4096, Dc = 1024;
constexpr int CHUNK   = 64;        // l-positions per thread (halo overhead 2/64)
constexpr int NCHUNK  = Lc / CHUNK;
constexpr int THREADS = Dc / 4;    // 256 threads; each owns 4 consecutive channels

__device__ __forceinline__ float gelu_exact(float v) {
  // approximate=False: 0.5*v*(1+erf(v/sqrt(2)))
  return 0.5f * v * (1.0f + erff(v * 0.70710678118654752440f));
}

union F4 { float4 v; float f[4]; };

__global__ __launch_bounds__(THREADS) void gated_dwconv_k3(
    const float* __restrict__ x, const float* __restrict__ w_conv,
    const float* __restrict__ w_gate, float* __restrict__ out) {
  __shared__ float lds_w[2 * 3 * Dc];  // 24 KB: [0,3072)=w_conv, [3072,6144)=w_gate
  const int t = threadIdx.x;

  // ---- CDNA5 async data mover: stage weight tables global -> LDS ----
#if __has_builtin(__builtin_amdgcn_global_load_async_to_lds_b128)
  {
    typedef int v4i __attribute__((vector_size(16)));
    typedef __attribute__((address_space(1))) v4i* g4_t;  // global (prints as __device__)
    typedef __attribute__((address_space(3))) v4i* l4_t;  // LDS
    #pragma unroll
    for (int p = 0; p < 3; ++p) {
      const int o = p * (THREADS * 4) + 4 * t;  // float4 stride over 3072 floats
      __builtin_amdgcn_global_load_async_to_lds_b128(
          (g4_t)(w_conv + o), (l4_t)(&lds_w[o]), 0, 0);
      __builtin_amdgcn_global_load_async_to_lds_b128(
          (g4_t)(w_gate + o), (l4_t)(&lds_w[3 * Dc + o]), 0, 0);
    }
  #if __has_builtin(__builtin_amdgcn_s_wait_asynccnt)
    __builtin_amdgcn_s_wait_asynccnt(0);
  #else
    asm volatile("s_wait_asynccnt 0x0" ::: "memory");
  #endif
  }
#else
  for (int i = t; i < 3 * Dc; i += THREADS) {
    lds_w[i]          = w_conv[i];
    lds_w[3 * Dc + i] = w_gate[i];
  }
#endif
  __syncthreads();

  const int b  = blockIdx.x / NCHUNK;
  const int l0 = (blockIdx.x % NCHUNK) * CHUNK;
  const int d0 = 4 * t;

  // Per-thread taps for 4 channels (from LDS; read once, live in VGPRs).
  float wc[3][4], wg[3][4];
  #pragma unroll
  for (int c = 0; c < 4; ++c) {
    #pragma unroll
    for (int k = 0; k < 3; ++k) {
      wc[k][c] = lds_w[(d0 + c) * 3 + k];
      wg[k][c] = lds_w[3 * Dc + (d0 + c) * 3 + k];
    }
  }

  const size_t base = (size_t)b * Lc * Dc + (size_t)d0;

  // Causal halo: l0 is 0 or a multiple of CHUNK (>=64), so either both halo
  // rows exist or both are the zero-padding.
  F4 xm2, xm1;
  xm2.v = make_float4(0.f, 0.f, 0.f, 0.f);
  xm1.v = xm2.v;
  if (l0 > 0) {
    xm2.v = *(const float4*)(x + base + (size_t)(l0 - 2) * Dc);
    xm1.v = *(const float4*)(x + base + (size_t)(l0 - 1) * Dc);
  }

  #pragma unroll 4
  for (int l = l0; l < l0 + CHUNK; ++l) {
    F4 xc, o;
    xc.v = *(const float4*)(x + base + (size_t)l * Dc);  // global_load_b128
    #pragma unroll
    for (int c = 0; c < 4; ++c) {
      float up = fmaf(xm2.f[c], wc[0][c], fmaf(xm1.f[c], wc[1][c], xc.f[c] * wc[2][c]));
      float gt = fmaf(xm2.f[c], wg[0][c], fmaf(xm1.f[c], wg[1][c], xc.f[c] * wg[2][c]));
      o.f[c] = up * gelu_exact(gt);
    }
    *(float4*)(out + base + (size_t)l * Dc) = o.v;     // global_store_b128
    xm2 = xm1;
    xm1 = xc;
  }
}

extern "C" void kernel_launch(void* const* d_in, const int* in_sizes, int n_in,
                              void* d_out, int out_size, void* d_ws, size_t ws_size,
                              hipStream_t stream) {
  const float* x      = (const float*)d_in[0];
  const float* w_conv = (const float*)d_in[1];
  const float* w_gate = (const float*)d_in[2];
  float* out = (float*)d_out;

  dim3 grid(Bc * NCHUNK);   // 512 blocks = 4096 waves
  dim3 block(THREADS);      // 256 threads = 8 wave32s
  gated_dwconv_k3<<<grid, block, 0, stream>>>(x, w_conv, w_gate, out);
}